// PsiNN0_84688165142617
// MI455X (gfx1250) — compile-verified
//
#include <hip/hip_runtime.h>
#include <math.h>

// ---------------------------------------------------------------------------
// Echo-state network for MI455X (gfx1250, wave32).
//   U = x @ Win^T          : WMMA f32 16x16x4 GEMM (parallel, exact f32)
//   h_t recurrence         : VALU matvec, W L2-resident, grid barrier / step
//   out_t = Wout @ h_t     : folded into recurrence (reads LDS-staged h)
// ---------------------------------------------------------------------------

typedef __attribute__((ext_vector_type(2))) float v2f;
typedef __attribute__((ext_vector_type(8))) float v8f;

#define SEQ    8192
#define INDIM  64
#define RES    2048
#define OUTDIM 128
#define ALPHA  0.5f

#define NWG 128          // workgroups in recurrence kernel (== OUTDIM)
#define TPB 256          // threads per workgroup (8 waves of 32)
#define ROWS_PER_WG (RES / NWG)   // 16 rows of h per WG, 16 lanes per row

// ---------------------------------------------------------------------------
// Kernel A: U[s,r] = sum_k x[s,k] * Win[r,k]   (8192 x 2048, K = 64)
// One wave computes one 16x16 tile with 16 chained V_WMMA_F32_16X16X4_F32.
// ---------------------------------------------------------------------------
__global__ __launch_bounds__(256) void esn_pregemm(const float* __restrict__ x,
                                                   const float* __restrict__ Win,
                                                   float* __restrict__ U) {
  const int lane = threadIdx.x & 31;
  const int wave = threadIdx.x >> 5;
  const int tile = blockIdx.x * 8 + wave;      // 512*128 tiles total
  const int RT = RES / 16;                     // 128 r-tiles
  const int sT = tile / RT;
  const int rT = tile % RT;
  const int lm = lane & 15;
  const int hi = lane >> 4;                    // 0 or 1 (upper K pair / upper M half)

  // A lane j: A[M=j&15][K = 2*(j>>4)+v], B lane j: B[K = 2*(j>>4)+v][N=j&15]
  const float* xr = x   + (size_t)(sT * 16 + lm) * INDIM + 2 * hi;
  const float* wr = Win + (size_t)(rT * 16 + lm) * INDIM + 2 * hi;

  v8f acc = {};
#pragma unroll
  for (int k = 0; k < INDIM; k += 4) {
    v2f a = *(const v2f*)(xr + k);
    v2f b = *(const v2f*)(wr + k);
    acc = __builtin_amdgcn_wmma_f32_16x16x4_f32(
        /*neg_a=*/false, a, /*neg_b=*/false, b,
        /*c_mod=*/(short)0, acc, /*reuse_a=*/false, /*reuse_b=*/false);
  }
  // D: VGPR r, lane j -> C[M = 8*(j>>4)+r][N = j&15]
  float* o = U + (size_t)(sT * 16 + hi * 8) * RES + rT * 16 + lm;
#pragma unroll
  for (int r = 0; r < 8; ++r) o[(size_t)r * RES] = acc[r];
}

// ---------------------------------------------------------------------------
// Init: zero h0 (double-buffer slot 0) and the grid-barrier state each call
// (workspace is NOT re-poisoned between graph replays -> must self-reset).
// ---------------------------------------------------------------------------
__global__ void esn_init(float* hbuf, unsigned* bar) {
  int i = blockIdx.x * blockDim.x + threadIdx.x;
  if (i < RES) hbuf[i] = 0.0f;
  if (i < 2)   bar[i] = 0u;
}

// ---------------------------------------------------------------------------
// Kernel B: sequential recurrence + fused output projection.
// ---------------------------------------------------------------------------
__global__ __launch_bounds__(TPB) void esn_recurrence(
    const float* __restrict__ x, const float* __restrict__ Win,
    const float* __restrict__ W, const float* __restrict__ Wout,
    const float* __restrict__ U, int useU,
    float* __restrict__ out, float* __restrict__ hbuf,
    unsigned* bar_cnt, unsigned* bar_gen) {
  __shared__ float hs[RES];     // full h vector staged per step (8 KB)
  __shared__ float red[TPB / 32];

  const int tid  = threadIdx.x;
  const int lane = tid & 31;
  const int wv   = tid >> 5;
  const int rowL = tid >> 4;                         // 0..15
  const int seg  = tid & 15;                         // 16 lanes per row
  const int rowG = blockIdx.x * ROWS_PER_WG + rowL;  // global h row
  const int oCol = blockIdx.x;                       // output column (NWG==OUTDIM)

  const float* wrow = W    + (size_t)rowG * RES + seg * 4;
  const float* winr = Win  + (size_t)rowG * INDIM;
  const float* wo   = Wout + (size_t)oCol * RES + tid * 8;

  for (int t = 0; t < SEQ; ++t) {
    const float* hprev = hbuf + (size_t)(t & 1) * RES;
    float*       hnext = hbuf + (size_t)((t & 1) ^ 1) * RES;

    // ---- stage h_{t-1} into LDS (512 float4, 2 per thread) ----
    for (int i = tid; i < RES / 4; i += TPB)
      ((float4*)hs)[i] = ((const float4*)hprev)[i];
    __syncthreads();

    // ---- fused projection for step t-1: out[t-1][oCol] = Wout[oCol]·h_{t-1} ----
    if (t > 0) {
      float p = 0.0f;
      const float* hp = hs + tid * 8;
#pragma unroll
      for (int j = 0; j < 8; ++j) p += wo[j] * hp[j];
#pragma unroll
      for (int m = 16; m; m >>= 1) p += __shfl_xor(p, m, 32);
      if (lane == 0) red[wv] = p;
      __syncthreads();
      if (tid == 0) {
        float s = 0.0f;
#pragma unroll
        for (int j = 0; j < TPB / 32; ++j) s += red[j];
        out[(size_t)(t - 1) * OUTDIM + oCol] = s;
      }
    }

    // ---- matvec: sum_k W[rowG][k] * h_{t-1}[k]  (float4, 16 lanes/row) ----
    float sum = 0.0f;
#pragma unroll 4
    for (int j = 0; j < RES / 64; ++j) {               // 32 iterations
      float4 w4 = *(const float4*)(wrow + j * 64);
      float4 h4 = *(const float4*)(hs + seg * 4 + j * 64);
      sum += w4.x * h4.x + w4.y * h4.y + w4.z * h4.z + w4.w * h4.w;
    }
#pragma unroll
    for (int m = 8; m; m >>= 1) sum += __shfl_xor(sum, m, 16);

    // ---- input drive u = (Win x_t)[rowG] ----
    float u = 0.0f;
    if (useU) {
      if (seg == 0) u = U[(size_t)t * RES + rowG];
    } else {
      const float* xt = x + (size_t)t * INDIM;
      float uu = 0.0f;
#pragma unroll
      for (int j = 0; j < 4; ++j) uu += winr[seg * 4 + j] * xt[seg * 4 + j];
#pragma unroll
      for (int m = 8; m; m >>= 1) uu += __shfl_xor(uu, m, 16);
      u = uu;
    }

    if (seg == 0) {
      float hn = (1.0f - ALPHA) * hs[rowG] + ALPHA * tanhf(u + sum);
      hnext[rowG] = hn;
    }

    // ---- grid barrier (generation == t+1, monotonic, deterministic) ----
    __syncthreads();
    if (tid == 0) {
      __threadfence();
      unsigned arrived = atomicAdd(bar_cnt, 1u);
      if (arrived == NWG - 1) {
        atomicExch(bar_cnt, 0u);        // reset before publishing generation
        __threadfence();
        atomicExch(bar_gen, (unsigned)(t + 1));
      } else {
        while (atomicAdd(bar_gen, 0u) < (unsigned)(t + 1))
          __builtin_amdgcn_s_sleep(2);
      }
      __threadfence();
    }
    __syncthreads();
  }

  // ---- final projection for t = SEQ-1 (h_{SEQ-1} lives in hbuf slot 0) ----
  const float* hlast = hbuf + (size_t)(SEQ & 1) * RES;   // == hbuf
  for (int i = tid; i < RES / 4; i += TPB)
    ((float4*)hs)[i] = ((const float4*)hlast)[i];
  __syncthreads();
  {
    float p = 0.0f;
    const float* hp = hs + tid * 8;
#pragma unroll
    for (int j = 0; j < 8; ++j) p += wo[j] * hp[j];
#pragma unroll
    for (int m = 16; m; m >>= 1) p += __shfl_xor(p, m, 32);
    if (lane == 0) red[wv] = p;
    __syncthreads();
    if (tid == 0) {
      float s = 0.0f;
#pragma unroll
      for (int j = 0; j < TPB / 32; ++j) s += red[j];
      out[(size_t)(SEQ - 1) * OUTDIM + oCol] = s;
    }
  }
}

// ---------------------------------------------------------------------------
extern "C" void kernel_launch(void* const* d_in, const int* in_sizes, int n_in,
                              void* d_out, int out_size, void* d_ws, size_t ws_size,
                              hipStream_t stream) {
  const float* x    = (const float*)d_in[0];   // [8192, 64]
  const float* Win  = (const float*)d_in[1];   // [2048, 64]
  const float* W    = (const float*)d_in[2];   // [2048, 2048]
  const float* Wout = (const float*)d_in[3];   // [128, 2048]
  float* out = (float*)d_out;                  // [8192, 128]

  // workspace layout: [barrier 256B][hbuf 2*RES f32][pad][U 8192*2048 f32]
  char* ws = (char*)d_ws;
  unsigned* bar  = (unsigned*)ws;
  float*    hbuf = (float*)(ws + 256);
  size_t uOff = 256 + (size_t)2 * RES * sizeof(float);
  uOff = (uOff + 255) & ~(size_t)255;
  float* U = (float*)(ws + uOff);
  const size_t need = uOff + (size_t)SEQ * RES * sizeof(float);
  const int useU = (ws_size >= need) ? 1 : 0;

  esn_init<<<(RES + 255) / 256, 256, 0, stream>>>(hbuf, bar);

  if (useU) {
    const int tiles = (SEQ / 16) * (RES / 16);     // 65536 tiles, 8 waves/block
    esn_pregemm<<<tiles / 8, 256, 0, stream>>>(x, Win, U);
  }

  esn_recurrence<<<NWG, TPB, 0, stream>>>(x, Win, W, Wout, U, useU,
                                          out, hbuf, bar, bar + 1);
}